// UnbiasedIWTrueKnowledgeDistillationLoss_20512763806282
// MI455X (gfx1250) — compile-verified
//
#include <hip/hip_runtime.h>
#include <hip/hip_bf16.h>

// Problem constants (from reference setup_inputs)
#define BB   8
#define CC   21          // new-model channels
#define OLD  16          // old-model channels
#define HWW  (512 * 512) // 262144 = 2^18
#define PIX  (BB * HWW)  // 2,097,152 pixels

#define HIST_BLOCKS_PER_SAMPLE 128
#define LOSS_BLOCKS 2048
#define LOSS_THREADS 256

typedef float v2f __attribute__((ext_vector_type(2)));
typedef float v8f __attribute__((ext_vector_type(8)));
typedef int   v4i __attribute__((ext_vector_type(4)));

// CDNA5 async global->LDS path (guarded: falls back to direct loads if the
// toolchain doesn't declare these builtins).
#if defined(__has_builtin)
#if __has_builtin(__builtin_amdgcn_global_load_async_to_lds_b128) && \
    __has_builtin(__builtin_amdgcn_s_wait_asynccnt)
#define HAVE_ASYNC_LDS 1
#endif
#endif
#ifndef HAVE_ASYNC_LDS
#define HAVE_ASYNC_LDS 0
#endif

#if HAVE_ASYNC_LDS
typedef __attribute__((address_space(1))) v4i* gvec_t;  // global int4*
typedef __attribute__((address_space(3))) v4i* lvec_t;  // LDS int4*
#endif

// ---------------- workspace layout ----------------
// [0, 1024)        : unsigned hist[8][32]   (21 bins used, padded)
// [1024, 1536)     : float weights[8][16]
// [2048, 2048+8KB) : float partials[LOSS_BLOCKS]

__global__ void zero_hist_kernel(unsigned* __restrict__ hist) {
    int i = threadIdx.x;
    if (i < BB * 32) hist[i] = 0u;
}

__device__ __forceinline__ void hist_bin4(float4 v, unsigned* lh) {
    float xs[4] = {v.x, v.y, v.z, v.w};
#pragma unroll
    for (int k = 0; k < 4; ++k) {
        float x = xs[k];
        if (x >= 0.0f && x <= 20.0f) {            // excludes NaN too
            int bin = (int)(x * (21.0f / 20.0f)); // == floor for x>=0
            bin = bin > 20 ? 20 : bin;
            atomicAdd(&lh[bin], 1u);
        }
    }
}

// Per-sample histogram of targets (ALPHA == 1.0 so t == targets).
// torch.histc semantics: bins=21, min=0, max=20, width=20/21,
// idx = clip(floor(x*21/20), 0, 20), values outside [0,20] excluded.
// Sample slice divides evenly: 128 blocks x 256 thr x 32 iters x 4 floats
// == 16*512*512, so every thread runs exactly HIST_ITERS iterations (EXEC
// stays all-ones, no bounds checks).
#define HIST_ITERS 32
__global__ void hist_kernel(const float* __restrict__ tgt, unsigned* __restrict__ hist) {
    __shared__ unsigned lh[CC];
    const int b   = blockIdx.x / HIST_BLOCKS_PER_SAMPLE;
    const int blk = blockIdx.x % HIST_BLOCKS_PER_SAMPLE;
    if (threadIdx.x < CC) lh[threadIdx.x] = 0u;
    __syncthreads();

    const float4* base    = (const float4*)(tgt + (size_t)b * OLD * HWW);
    const unsigned stride = HIST_BLOCKS_PER_SAMPLE * 256;   // float4 stride
    const unsigned i0     = blk * 256u + threadIdx.x;

#if HAVE_ASYNC_LDS
    // Double-buffered async staging: the async DMA engine streams the next
    // 4KB tile into LDS (tracked by ASYNCcnt) while we histogram the current
    // tile. Each thread self-consumes its own 16B slot -> only s_wait_asynccnt
    // is needed, no barrier.
    __shared__ __align__(16) char stage[2][256 * 16];

    // prologue: issue tile 0
    __builtin_amdgcn_global_load_async_to_lds_b128(
        (gvec_t)(base + i0),
        (lvec_t)&stage[0][threadIdx.x * 16], 0, 0);
#pragma unroll 1
    for (int k = 0; k < HIST_ITERS - 1; ++k) {
        __builtin_amdgcn_global_load_async_to_lds_b128(
            (gvec_t)(base + i0 + (unsigned)(k + 1) * stride),
            (lvec_t)&stage[(k + 1) & 1][threadIdx.x * 16], 0, 0);
        __builtin_amdgcn_s_wait_asynccnt(1);   // current tile landed
        float4 v = *(const float4*)&stage[k & 1][threadIdx.x * 16];
        hist_bin4(v, lh);
    }
    __builtin_amdgcn_s_wait_asynccnt(0);
    {
        float4 v = *(const float4*)&stage[(HIST_ITERS - 1) & 1][threadIdx.x * 16];
        hist_bin4(v, lh);
    }
#else
#pragma unroll 1
    for (int k = 0; k < HIST_ITERS; ++k) {
        float4 v = base[i0 + (unsigned)k * stride];
        hist_bin4(v, lh);
    }
#endif

    __syncthreads();
    if (threadIdx.x < CC) atomicAdd(&hist[b * 32 + threadIdx.x], lh[threadIdx.x]);
}

// weights[b][c] = (total/h[c])^0.1 with hist==0 -> 1, hist[0] += sum(hist[16:21]).
__global__ void weights_kernel(const unsigned* __restrict__ hist, float* __restrict__ weights) {
    int t = threadIdx.x;
    if (t >= BB * OLD) return;
    int b = t / OLD, c = t % OLD;
    float h[CC];
#pragma unroll
    for (int j = 0; j < CC; ++j) {
        unsigned v = hist[b * 32 + j];
        h[j] = (v == 0u) ? 1.0f : (float)v;
    }
    float temp = 0.0f;
#pragma unroll
    for (int j = OLD; j < CC; ++j) temp += h[j];
    h[0] += temp;
    float total = 0.0f;
#pragma unroll
    for (int j = 0; j < OLD; ++j) total += h[j];
    weights[t] = powf(total / h[c], 0.1f);
}

// Fused per-pixel loss; deterministic per-block partial sums.
__global__ void loss_kernel(const float* __restrict__ inp, const float* __restrict__ tgt,
                            const float* __restrict__ weights, float* __restrict__ partials) {
    const unsigned stride = LOSS_BLOCKS * LOSS_THREADS; // 524288, PIX/stride == 4 exactly
    unsigned tid = blockIdx.x * blockDim.x + threadIdx.x;
    float acc = 0.0f;

    for (unsigned p = tid; p < PIX; p += stride) {
        unsigned b = p >> 18;           // HWW == 2^18
        unsigned q = p & (HWW - 1);
        const float* xin = inp + (size_t)b * CC * HWW + q;
        const float* tin = tgt + (size_t)b * OLD * HWW + q;

        // WGP-scope prefetch of next iteration's base lines (locality 3).
        unsigned p2 = p + stride;
        if (p2 < PIX) {
            unsigned b2 = p2 >> 18, q2 = p2 & (HWW - 1);
            __builtin_prefetch(inp + (size_t)b2 * CC * HWW + q2, 0, 3);
            __builtin_prefetch(tgt + (size_t)b2 * OLD * HWW + q2, 0, 3);
        }

        float x[CC], t[OLD];
#pragma unroll
        for (int c = 0; c < CC; ++c) x[c] = xin[(size_t)c * HWW];
#pragma unroll
        for (int c = 0; c < OLD; ++c) t[c] = tin[(size_t)c * HWW];

        // den = logsumexp over all 21 channels
        float m = x[0];
#pragma unroll
        for (int c = 1; c < CC; ++c) m = fmaxf(m, x[c]);
        float s = 0.0f;
#pragma unroll
        for (int c = 0; c < CC; ++c) s += __expf(x[c] - m);
        float den = m + __logf(s);

        // background logsumexp over channels {0, 16..20}
        float mb = x[0];
#pragma unroll
        for (int c = OLD; c < CC; ++c) mb = fmaxf(mb, x[c]);
        float sb = __expf(x[0] - mb);
#pragma unroll
        for (int c = OLD; c < CC; ++c) sb += __expf(x[c] - mb);
        float lse_bkg = mb + __logf(sb);

        // softmax over 16 target channels (in place)
        float mt = t[0];
#pragma unroll
        for (int c = 1; c < OLD; ++c) mt = fmaxf(mt, t[c]);
        float st = 0.0f;
#pragma unroll
        for (int c = 0; c < OLD; ++c) { t[c] = __expf(t[c] - mt); st += t[c]; }
        float inv = 1.0f / st;

        const float* w = weights + b * OLD;
        float lp = w[0] * t[0] * inv * (lse_bkg - den);
#pragma unroll
        for (int c = 1; c < OLD; ++c) lp += w[c] * (t[c] * inv) * (x[c] - den);
        acc += lp;
    }

    // ---- wave32 reduction via matrix engine ----
    // A(16x4): vgpr0=acc, vgpr1=0 -> A[m,0]=acc(lane m), A[m,2]=acc(lane m+16).
    // B = ones(4x16) -> D[m,n] = acc(m) + acc(m+16)  (uniform across n).
    // Per-lane sum of the 8 D VGPRs covers M=0..7 (lanes<16) or M=8..15
    // (lanes>=16); one shfl_xor(16) completes the 32-lane sum. EXEC all-ones.
    v2f a;   a[0] = acc;  a[1] = 0.0f;
    v2f one; one[0] = 1.0f; one[1] = 1.0f;
    v8f cz = {};
    v8f d = __builtin_amdgcn_wmma_f32_16x16x4_f32(false, a, false, one, (short)0, cz, false, false);
    float h8 = d[0] + d[1] + d[2] + d[3] + d[4] + d[5] + d[6] + d[7];
    float wsum = h8 + __shfl_xor(h8, 16, 32);

    __shared__ float sw[LOSS_THREADS / 32];
    int lane = threadIdx.x & 31, wv = threadIdx.x >> 5;
    if (lane == 0) sw[wv] = wsum;
    __syncthreads();
    if (threadIdx.x == 0) {
        float bs = 0.0f;
#pragma unroll
        for (int i = 0; i < LOSS_THREADS / 32; ++i) bs += sw[i];
        partials[blockIdx.x] = bs;
    }
}

__global__ void final_kernel(const float* __restrict__ partials, float* __restrict__ out) {
    __shared__ float s[256];
    float v = 0.0f;
    for (int i = threadIdx.x; i < LOSS_BLOCKS; i += 256) v += partials[i];
    s[threadIdx.x] = v;
    __syncthreads();
    for (int off = 128; off > 0; off >>= 1) {
        if (threadIdx.x < off) s[threadIdx.x] += s[threadIdx.x + off];
        __syncthreads();
    }
    if (threadIdx.x == 0) out[0] = -s[0] / (float)PIX;
}

extern "C" void kernel_launch(void* const* d_in, const int* in_sizes, int n_in,
                              void* d_out, int out_size, void* d_ws, size_t ws_size,
                              hipStream_t stream) {
    (void)in_sizes; (void)n_in; (void)out_size; (void)ws_size;
    const float* inp = (const float*)d_in[0];   // [8,21,512,512] fp32
    const float* tgt = (const float*)d_in[1];   // [8,16,512,512] fp32
    unsigned* hist    = (unsigned*)d_ws;
    float*    weights = (float*)((char*)d_ws + 1024);
    float*    partials= (float*)((char*)d_ws + 2048);
    float*    out     = (float*)d_out;

    hipLaunchKernelGGL(zero_hist_kernel, dim3(1), dim3(256), 0, stream, hist);
    hipLaunchKernelGGL(hist_kernel, dim3(BB * HIST_BLOCKS_PER_SAMPLE), dim3(256), 0, stream, tgt, hist);
    hipLaunchKernelGGL(weights_kernel, dim3(1), dim3(128), 0, stream, hist, weights);
    hipLaunchKernelGGL(loss_kernel, dim3(LOSS_BLOCKS), dim3(LOSS_THREADS), 0, stream, inp, tgt, weights, partials);
    hipLaunchKernelGGL(final_kernel, dim3(1), dim3(256), 0, stream, partials, out);
}